// StudentModel_43654047596825
// MI455X (gfx1250) — compile-verified
//
#include <hip/hip_runtime.h>
#include <hip/hip_bf16.h>
#include <math.h>

#define HEADS 3
#define HID   256
#define NEG_SLOPE 0.2f

typedef __attribute__((ext_vector_type(2))) float v2f;
typedef __attribute__((ext_vector_type(8))) float v8f;

// ---------------------------------------------------------------------------
// float atomic max via integer atomics (sign-aware bit trick)
// ---------------------------------------------------------------------------
__device__ __forceinline__ void atomicMaxF(float* addr, float val) {
    if (val >= 0.0f) {
        atomicMax((int*)addr, __float_as_int(val));
    } else {
        atomicMin((unsigned int*)addr, (unsigned int)__float_as_int(val));
    }
}

// ---------------------------------------------------------------------------
// Zero-pad a weight matrix W[K, M] -> Wp[Kpad, Mpad]
// ---------------------------------------------------------------------------
__global__ void gat_pad_weights(const float* __restrict__ W, int K, int M,
                                float* __restrict__ Wp, int Kpad, int Mpad) {
    long i = (long)blockIdx.x * blockDim.x + threadIdx.x;
    if (i >= (long)Kpad * Mpad) return;
    int r = (int)(i / Mpad);
    int c = (int)(i - (long)r * Mpad);
    Wp[i] = (r < K && c < M) ? W[(size_t)r * M + c] : 0.0f;
}

// ---------------------------------------------------------------------------
// GEMM: C[N, Mstore] = A[N, K] @ B[Kpad, Mpad] (+ bias)
// fp32 via V_WMMA_F32_16X16X4_F32. B is pre-padded: Kpad % 4 == 0,
// Mpad % 16 == 0, so the inner K loop is completely branch-free.
// One workgroup = 16 waves = 512 threads handles a 16-row stripe;
// the A tile is staged (zero-padded) in LDS and reused by all column tiles.
// ---------------------------------------------------------------------------
__global__ __launch_bounds__(512)
void gat_gemm16_wmma(const float* __restrict__ A, int N, int K, int Kpad,
                     const float* __restrict__ B, int Mpad, int Mstore,
                     const float* __restrict__ bias,
                     float* __restrict__ C, int ldC) {
    __shared__ float As[16][257];   // Kpad <= 256; +1 stride pad -> conflict-free

    const int row0 = blockIdx.x << 4;
    const int tid  = threadIdx.x;

    // Stage A tile (coalesced along K); zero-pad out-of-range rows / K tail.
    for (int i = tid; i < 16 * Kpad; i += 512) {
        int r  = i / Kpad;
        int k  = i - r * Kpad;
        int gr = row0 + r;
        As[r][k] = (gr < N && k < K) ? A[(size_t)gr * K + k] : 0.0f;
    }
    __syncthreads();

    const int wave = tid >> 5;
    const int lane = tid & 31;
    const int half = lane >> 4;   // 0: lanes 0-15, 1: lanes 16-31
    const int rr   = lane & 15;

    const int ntiles = Mpad >> 4;
    for (int t = wave; t < ntiles; t += 16) {
        const int col = (t << 4) + rr;
        // Per-lane base pointer into B for this column; advance by 4 rows/step.
        const float* bp = B + (size_t)(2 * half) * Mpad + col;

        v8f acc = {};
#pragma unroll 4
        for (int k = 0; k < Kpad; k += 4) {
            v2f a, b;
            // A 16x4: lane rr holds row rr; half-wave selects K pair {0,1}/{2,3}
            a.x = As[rr][k + 2 * half];
            a.y = As[rr][k + 2 * half + 1];
            // B 4x16: VGPR0 = rows {0,2}, VGPR1 = rows {1,3}, striped over lanes
            b.x = bp[0];
            b.y = bp[Mpad];
            bp += (size_t)4 * Mpad;
            acc = __builtin_amdgcn_wmma_f32_16x16x4_f32(
                false, a, false, b, (short)0, acc, false, false);
        }

        if (col < Mstore) {
            const float bv = bias ? bias[col] : 0.0f;
#pragma unroll
            for (int j = 0; j < 8; ++j) {
                const int row = row0 + j + 8 * half;   // C: vgpr j -> rows j / j+8
                if (row < N) C[(size_t)row * ldC + col] = acc[j] + bv;
            }
        }
    }
}

// ---------------------------------------------------------------------------
// Per-node attention logits: alpha_s[n,h] = <h[n,h,:], a_src[h,:]>, same for dst
// One wave per (node, head); coalesced 32-lane strided dot + butterfly reduce.
// ---------------------------------------------------------------------------
__global__ __launch_bounds__(256)
void gat_alpha_kernel(const float* __restrict__ h,
                      const float* __restrict__ a_src,
                      const float* __restrict__ a_dst,
                      float* __restrict__ as_out, float* __restrict__ ad_out,
                      int N) {
    const int gw   = blockIdx.x * 8 + (threadIdx.x >> 5);
    const int lane = threadIdx.x & 31;
    if (gw >= N * HEADS) return;
    const int n  = gw / HEADS;
    const int hd = gw - n * HEADS;

    const float* hp = h + (size_t)n * (HEADS * HID) + hd * HID;
    const float* sp = a_src + hd * HID;
    const float* dp = a_dst + hd * HID;

    float ss = 0.0f, sd = 0.0f;
#pragma unroll
    for (int c = lane; c < HID; c += 32) {
        const float v = hp[c];
        ss += v * sp[c];
        sd += v * dp[c];
    }
#pragma unroll
    for (int off = 16; off > 0; off >>= 1) {
        ss += __shfl_xor(ss, off, 32);
        sd += __shfl_xor(sd, off, 32);
    }
    if (lane == 0) { as_out[gw] = ss; ad_out[gw] = sd; }
}

// ---------------------------------------------------------------------------
// Fill helper
// ---------------------------------------------------------------------------
__global__ void gat_fill_kernel(float* __restrict__ p, long n, float v) {
    long i = (long)blockIdx.x * blockDim.x + threadIdx.x;
    if (i < n) p[i] = v;
}

// ---------------------------------------------------------------------------
// Edge pass 1: e = leaky_relu(as[src]+ad[dst]); store e; atomic segment max
// ---------------------------------------------------------------------------
__global__ void gat_edge_max_kernel(const int* __restrict__ ei, int E, int N,
                                    const float* __restrict__ as_,
                                    const float* __restrict__ ad_,
                                    float* __restrict__ ebuf,
                                    float* __restrict__ m) {
    const int e = blockIdx.x * blockDim.x + threadIdx.x;
    if (e >= E + N) return;
    int s, d;
    if (e < E) { s = ei[e]; d = ei[E + e]; } else { s = e - E; d = s; }
#pragma unroll
    for (int hd = 0; hd < HEADS; ++hd) {
        float v = as_[s * HEADS + hd] + ad_[d * HEADS + hd];
        v = (v > 0.0f) ? v : NEG_SLOPE * v;
        ebuf[(size_t)e * HEADS + hd] = v;
        atomicMaxF(&m[d * HEADS + hd], v);
    }
}

// ---------------------------------------------------------------------------
// Edge pass 2: ex = exp(e - m[dst]); overwrite ebuf with ex; atomic denom sum
// ---------------------------------------------------------------------------
__global__ void gat_edge_exp_kernel(const int* __restrict__ ei, int E, int N,
                                    float* __restrict__ ebuf,
                                    const float* __restrict__ m,
                                    float* __restrict__ denom) {
    const int e = blockIdx.x * blockDim.x + threadIdx.x;
    if (e >= E + N) return;
    int d;
    if (e < E) { d = ei[E + e]; } else { d = e - E; }
#pragma unroll
    for (int hd = 0; hd < HEADS; ++hd) {
        const float ex = expf(ebuf[(size_t)e * HEADS + hd] - m[d * HEADS + hd]);
        ebuf[(size_t)e * HEADS + hd] = ex;
        atomicAdd(&denom[d * HEADS + hd], ex);
    }
}

// ---------------------------------------------------------------------------
// Edge pass 3: fused head-mean scatter:
//   agg[dst, c] += sum_h alpha[e,h] * h[src, h, c]
// One 256-thread block per edge; thread c is fully coalesced on h and agg.
// ---------------------------------------------------------------------------
__global__ __launch_bounds__(256)
void gat_aggregate_kernel(const int* __restrict__ ei, int E, int N,
                          const float* __restrict__ h,
                          const float* __restrict__ ebuf,
                          const float* __restrict__ denom,
                          float* __restrict__ agg) {
    const int e   = blockIdx.x;
    const int tid = threadIdx.x;
    __shared__ int   sd[2];
    __shared__ float al[HEADS];

    if (tid == 0) {
        int s, d;
        if (e < E) { s = ei[e]; d = ei[E + e]; } else { s = e - E; d = s; }
        sd[0] = s; sd[1] = d;
    }
    __syncthreads();
    const int s = sd[0];
    const int d = sd[1];
    if (tid < HEADS) {
        al[tid] = ebuf[(size_t)e * HEADS + tid] /
                  (denom[d * HEADS + tid] + 1e-16f);
    }
    __syncthreads();

    const float* hp = h + (size_t)s * (HEADS * HID);
    const float v = al[0] * hp[tid]
                  + al[1] * hp[HID + tid]
                  + al[2] * hp[2 * HID + tid];
    atomicAdd(&agg[(size_t)d * HID + tid], v);
}

// ---------------------------------------------------------------------------
// Finalize: feat = elu(agg/HEADS + bias)
// ---------------------------------------------------------------------------
__global__ void gat_finalize_kernel(float* __restrict__ feat,
                                    const float* __restrict__ bias, long n) {
    long i = (long)blockIdx.x * blockDim.x + threadIdx.x;
    if (i >= n) return;
    const int c = (int)(i & (HID - 1));
    float v = feat[i] * (1.0f / HEADS) + bias[c];
    feat[i] = (v > 0.0f) ? v : (expf(v) - 1.0f);
}

// ---------------------------------------------------------------------------
// Host-side orchestration
// ---------------------------------------------------------------------------
extern "C" void kernel_launch(void* const* d_in, const int* in_sizes, int n_in,
                              void* d_out, int out_size, void* d_ws, size_t ws_size,
                              hipStream_t stream) {
    (void)n_in; (void)out_size; (void)ws_size;

    const float* x  = (const float*)d_in[0];
    const int*   ei = (const int*)d_in[1];

    const float* W[3]  = { (const float*)d_in[2], (const float*)d_in[6],  (const float*)d_in[10] };
    const float* AS[3] = { (const float*)d_in[3], (const float*)d_in[7],  (const float*)d_in[11] };
    const float* AD[3] = { (const float*)d_in[4], (const float*)d_in[8],  (const float*)d_in[12] };
    const float* BV[3] = { (const float*)d_in[5], (const float*)d_in[9],  (const float*)d_in[13] };
    const float* fw = (const float*)d_in[14];
    const float* fb = (const float*)d_in[15];

    const int F_IN = in_sizes[2] / (HEADS * HID);   // 50
    const int N    = in_sizes[0] / F_IN;            // 50000
    const int E    = in_sizes[1] / 2;               // 400000
    const int CLS  = in_sizes[15];                  // 121
    const int Etot = E + N;
    const int HC   = HEADS * HID;                   // 768

    const int K1pad   = (F_IN + 3) & ~3;            // 52
    const int CLSpad  = (CLS + 15) & ~15;           // 128

    // Workspace layout (fp32)
    float* ws    = (float*)d_ws;
    float* h_buf = ws;  ws += (size_t)N * HC;        // [N, 768]
    float* feat  = ws;  ws += (size_t)N * HID;       // [N, 256] (GEMM input / agg target)
    float* ebuf  = ws;  ws += (size_t)Etot * HEADS;  // per-edge logits/weights
    float* mbuf  = ws;  ws += (size_t)N * HEADS;
    float* dbuf  = ws;  ws += (size_t)N * HEADS;
    float* asb   = ws;  ws += (size_t)N * HEADS;
    float* adb   = ws;  ws += (size_t)N * HEADS;
    float* w1pad = ws;  ws += (size_t)K1pad * HC;    // [52, 768]
    float* fwpad = ws;  ws += (size_t)HID * CLSpad;  // [256, 128]

    // Pre-pad weights so the GEMM inner loop is branch-free.
    {
        long n1 = (long)K1pad * HC;
        gat_pad_weights<<<(n1 + 255) / 256, 256, 0, stream>>>(W[0], F_IN, HC, w1pad, K1pad, HC);
        long n2 = (long)HID * CLSpad;
        gat_pad_weights<<<(n2 + 255) / 256, 256, 0, stream>>>(fw, HID, CLS, fwpad, HID, CLSpad);
    }

    const int gemm_grid  = (N + 15) / 16;
    const int alpha_grid = (N * HEADS + 7) / 8;
    const int edge_grid  = (Etot + 255) / 256;

    const float* layer_in = x;
    int K = F_IN;

    for (int l = 0; l < 3; ++l) {
        const int Kpad = (K + 3) & ~3;
        const float* Bmat = (l == 0) ? w1pad : W[l];

        // h = in @ W   [N, 768]
        gat_gemm16_wmma<<<gemm_grid, 512, 0, stream>>>(
            layer_in, N, K, Kpad, Bmat, HC, HC, nullptr, h_buf, HC);

        // per-node attention logits
        gat_alpha_kernel<<<alpha_grid, 256, 0, stream>>>(
            h_buf, AS[l], AD[l], asb, adb, N);

        // init segment state + aggregation target (layer_in now dead)
        gat_fill_kernel<<<((long)N * HID + 255) / 256, 256, 0, stream>>>(feat, (long)N * HID, 0.0f);
        gat_fill_kernel<<<((long)N * HEADS + 255) / 256, 256, 0, stream>>>(dbuf, (long)N * HEADS, 0.0f);
        gat_fill_kernel<<<((long)N * HEADS + 255) / 256, 256, 0, stream>>>(mbuf, (long)N * HEADS, -INFINITY);

        // segment softmax (3 passes) + fused head-mean scatter
        gat_edge_max_kernel<<<edge_grid, 256, 0, stream>>>(ei, E, N, asb, adb, ebuf, mbuf);
        gat_edge_exp_kernel<<<edge_grid, 256, 0, stream>>>(ei, E, N, ebuf, mbuf, dbuf);
        gat_aggregate_kernel<<<Etot, 256, 0, stream>>>(ei, E, N, h_buf, ebuf, dbuf, feat);

        // feat = elu(feat/3 + b)
        gat_finalize_kernel<<<((long)N * HID + 255) / 256, 256, 0, stream>>>(feat, BV[l], (long)N * HID);

        layer_in = feat;
        K = HID;
    }

    // classifier: out = feat @ fw + fb   [N, 121]
    gat_gemm16_wmma<<<gemm_grid, 512, 0, stream>>>(
        feat, N, HID, HID, fwpad, CLSpad, CLS, fb, (float*)d_out, CLS);
}